// BahdanauAttention_37976100831903
// MI455X (gfx1250) — compile-verified
//
#include <hip/hip_runtime.h>

typedef __attribute__((ext_vector_type(2))) float v2f;
typedef __attribute__((ext_vector_type(4))) float v4f;
typedef __attribute__((ext_vector_type(8))) float v8f;

#define B_ 4
#define T_ 256
#define S_ 512
#define H_ 512

__device__ __forceinline__ float fast_tanh(float x) {
#if __has_builtin(__builtin_amdgcn_tanhf)
  return __builtin_amdgcn_tanhf(x);   // v_tanh_f32 on gfx1250
#else
  // numerically safe: large +x -> 1, large -x -> -1
  float e = __expf(2.0f * x);
  return 1.0f - 2.0f / (e + 1.0f);
#endif
}

// out[m][n] = sum_h X[m][h] * W[n][h] + bias[n]
// X: M x H row-major; W: N(=H_) x H row-major (so B-matrix = W^T; frags contiguous in h)
// One wave (32 lanes) computes one 16x16 output tile via V_WMMA_F32_16X16X4_F32.
__global__ void proj_gemm(const float* __restrict__ X, const float* __restrict__ W,
                          const float* __restrict__ bias, float* __restrict__ out,
                          int M) {
  const int lane = threadIdx.x & 31;
  const int wavesPerBlock = blockDim.x >> 5;
  const int gw = blockIdx.x * wavesPerBlock + (threadIdx.x >> 5);
  const int tilesN = H_ / 16;
  const int tileM = gw / tilesN;
  const int tileN = gw % tilesN;
  if (tileM * 16 >= M) return;                 // wave-uniform; EXEC stays all-1s
  const int half = lane >> 4;                  // 0: K=0,1   1: K=2,3
  const int idx  = lane & 15;

  const float* arow = X + (size_t)(tileM * 16 + idx) * H_;
  const float* brow = W + (size_t)(tileN * 16 + idx) * H_;

  v8f acc = {};
  for (int kk = 0; kk < H_; kk += 4) {
    v2f a = *(const v2f*)(arow + kk + 2 * half);
    v2f b = *(const v2f*)(brow + kk + 2 * half);
    acc = __builtin_amdgcn_wmma_f32_16x16x4_f32(false, a, false, b,
                                                (short)0, acc, false, false);
  }
  const float bv = bias[tileN * 16 + idx];
  float* orow = out + (size_t)(tileM * 16 + 8 * half) * H_ + tileN * 16 + idx;
#pragma unroll
  for (int r = 0; r < 8; ++r)
    orow[(size_t)r * H_] = acc[r] + bv;
}

// One block per (b,t): score[s] = sum_h V[h]*tanh(q[bt,h]+k[b,s,h]) + bV, softmax over s.
// Inner loop vectorized 4-wide: per 4 h-values -> 2x global b128 (two s rows),
// 2x ds b128 (q, V broadcast), 8 tanh chains. v_tanh_f32 (1/4 rate) is the issue floor.
__global__ void score_softmax(const float* __restrict__ q, const float* __restrict__ k,
                              const float* __restrict__ V, const float* __restrict__ bVp,
                              float* __restrict__ attn) {
  __shared__ v4f qs4[H_ / 4];
  __shared__ v4f vs4[H_ / 4];
  __shared__ float red[256];
  const int bt  = blockIdx.x;          // 0 .. B*T-1
  const int b   = bt / T_;
  const int tid = threadIdx.x;         // 256 threads

  {
    const v4f* qg = (const v4f*)(q + (size_t)bt * H_);
    const v4f* vg = (const v4f*)V;
    for (int h4 = tid; h4 < H_ / 4; h4 += 256) {
      qs4[h4] = qg[h4];
      vs4[h4] = vg[h4];
    }
  }
  __syncthreads();

  const float bV = bVp[0];
  const float* krow0 = k + ((size_t)b * S_ + tid) * H_;
  const float* krow1 = krow0 + (size_t)256 * H_;

  float acc0 = 0.0f, acc1 = 0.0f;
#pragma unroll 2
  for (int h4 = 0; h4 < H_ / 4; ++h4) {
    const v4f kv0 = *(const v4f*)(krow0 + 4 * h4);
    const v4f kv1 = *(const v4f*)(krow1 + 4 * h4);
    const v4f qv  = qs4[h4];
    const v4f vv  = vs4[h4];
    acc0 = fmaf(vv.x, fast_tanh(qv.x + kv0.x), acc0);
    acc0 = fmaf(vv.y, fast_tanh(qv.y + kv0.y), acc0);
    acc0 = fmaf(vv.z, fast_tanh(qv.z + kv0.z), acc0);
    acc0 = fmaf(vv.w, fast_tanh(qv.w + kv0.w), acc0);
    acc1 = fmaf(vv.x, fast_tanh(qv.x + kv1.x), acc1);
    acc1 = fmaf(vv.y, fast_tanh(qv.y + kv1.y), acc1);
    acc1 = fmaf(vv.z, fast_tanh(qv.z + kv1.z), acc1);
    acc1 = fmaf(vv.w, fast_tanh(qv.w + kv1.w), acc1);
  }
  const float sc0 = acc0 + bV;
  const float sc1 = acc1 + bV;

  // block max
  red[tid] = fmaxf(sc0, sc1);
  __syncthreads();
  for (int off = 128; off > 0; off >>= 1) {
    if (tid < off) red[tid] = fmaxf(red[tid], red[tid + off]);
    __syncthreads();
  }
  const float mx = red[0];
  __syncthreads();

  const float e0 = __expf(sc0 - mx);
  const float e1 = __expf(sc1 - mx);
  red[tid] = e0 + e1;
  __syncthreads();
  for (int off = 128; off > 0; off >>= 1) {
    if (tid < off) red[tid] += red[tid + off];
    __syncthreads();
  }
  const float inv = 1.0f / red[0];

  attn[(size_t)bt * S_ + tid]       = e0 * inv;
  attn[(size_t)bt * S_ + tid + 256] = e1 * inv;
}

// context[b,t,n] = sum_s attn[b,t,s] * enc[b,s,n]   (A: T x S row-major, B: S x H row-major)
__global__ void context_gemm(const float* __restrict__ attn, const float* __restrict__ enc,
                             float* __restrict__ out) {
  const int lane = threadIdx.x & 31;
  const int wavesPerBlock = blockDim.x >> 5;
  const int gw = blockIdx.x * wavesPerBlock + (threadIdx.x >> 5);
  const int tilesN = H_ / 16;                   // 32
  const int tilesPerBatch = (T_ / 16) * tilesN; // 512
  const int b  = gw / tilesPerBatch;
  const int t  = gw % tilesPerBatch;
  const int tileM = t / tilesN;
  const int tileN = t % tilesN;
  const int half = lane >> 4;
  const int idx  = lane & 15;

  const float* arow  = attn + ((size_t)b * T_ + tileM * 16 + idx) * S_;
  const float* bbase = enc + (size_t)b * S_ * H_ + tileN * 16 + idx;

  v8f acc = {};
  for (int kk = 0; kk < S_; kk += 4) {
    v2f a = *(const v2f*)(arow + kk + 2 * half);
    v2f bf;
    bf.x = bbase[(size_t)(kk + 2 * half)     * H_];
    bf.y = bbase[(size_t)(kk + 2 * half + 1) * H_];
    acc = __builtin_amdgcn_wmma_f32_16x16x4_f32(false, a, false, bf,
                                                (short)0, acc, false, false);
  }
  float* orow = out + ((size_t)b * T_ + tileM * 16 + 8 * half) * H_ + tileN * 16 + idx;
#pragma unroll
  for (int r = 0; r < 8; ++r)
    orow[(size_t)r * H_] = acc[r];
}

extern "C" void kernel_launch(void* const* d_in, const int* in_sizes, int n_in,
                              void* d_out, int out_size, void* d_ws, size_t ws_size,
                              hipStream_t stream) {
  const float* dec = (const float*)d_in[0];   // (B,T,H)
  const float* enc = (const float*)d_in[1];   // (B,S,H)
  const float* W1  = (const float*)d_in[2];   // (H,H)
  const float* b1  = (const float*)d_in[3];   // (H)
  const float* W2  = (const float*)d_in[4];   // (H,H)
  const float* b2  = (const float*)d_in[5];   // (H)
  const float* V   = (const float*)d_in[6];   // (H)
  const float* bV  = (const float*)d_in[7];   // scalar
  float* out = (float*)d_out;                 // (B,T,H)

  float* q    = (float*)d_ws;                 // B*T*H floats (2 MB)
  float* kbuf = q + (size_t)B_ * T_ * H_;     // B*S*H floats (4 MB)
  float* attn = kbuf + (size_t)B_ * S_ * H_;  // B*T*S floats (2 MB)

  // q projection: M=B*T=1024 -> 64*32=2048 tiles, 8 waves/block -> 256 blocks
  proj_gemm<<<256, 256, 0, stream>>>(dec, W1, b1, q, B_ * T_);
  // k projection: M=B*S=2048 -> 128*32=4096 tiles -> 512 blocks
  proj_gemm<<<512, 256, 0, stream>>>(enc, W2, b2, kbuf, B_ * S_);
  // scores + softmax: one block per (b,t)
  score_softmax<<<B_ * T_, 256, 0, stream>>>(q, kbuf, V, bV, attn);
  // context: 4 * 512 = 2048 tiles -> 256 blocks
  context_gemm<<<256, 256, 0, stream>>>(attn, enc, out);
}